// MlpAttentionLayer_82386062672079
// MI455X (gfx1250) — compile-verified
//
#include <hip/hip_runtime.h>
#include <math.h>

typedef __attribute__((ext_vector_type(2))) float v2f;
typedef __attribute__((ext_vector_type(4))) float v4f;
typedef __attribute__((ext_vector_type(8))) float v8f;
typedef __attribute__((ext_vector_type(4))) unsigned int u32x4;
typedef __attribute__((ext_vector_type(8))) int i32x8;
typedef __attribute__((ext_vector_type(4))) int i32x4;

#define BN_EPS  1e-3f
#define B_      2048
#define T_      200
#define D_      128
#define NT      13            // ceil(200/16) row tiles per batch
#define XPITCH  132           // padded row pitch (floats) -> bank-conflict-free
#define NWAVE   4
#define TILEF   (16 * XPITCH) // 2112 floats per X tile buffer

#if __has_builtin(__builtin_amdgcn_tensor_load_to_lds) && \
    __has_builtin(__builtin_amdgcn_s_wait_tensorcnt)
#define USE_TDM 1
#else
#define USE_TDM 0
#endif

#if __has_builtin(__builtin_amdgcn_sched_barrier)
#define SCHED_FENCE() __builtin_amdgcn_sched_barrier(0)
#else
#define SCHED_FENCE()
#endif

// ---- workspace layout (float offsets) ----
#define WS_ASW  0             // 16384 : A (W1'a+W1'b) pre-swizzled to WMMA B-operand order
#define WS_C    16384         // 16384 : C (W1'c-W1'b) row-major
#define WS_B1   32768         // 128   : folded bias b1'
#define WS_V    32896         // 128   : folded v = s2*W2
#define WS_C2   33024         // 1     : folded scalar c2

// ---- dynamic LDS layout (float offsets) ----
#define SM_W    0             // 16384 : swizzled weights
#define SM_QC   16384         // 128   : qc[b] (q@C + b1')
#define SM_V    16512         // 128   : v
#define SM_OUT  16640         // 128   : output accumulator
#define SM_C2   16768         // 1 (+pad)
#define SM_X    16784         // 4 waves * 2 * TILEF double-buffered X staging
#define SM_TOTAL (SM_X + NWAVE * 2 * TILEF)     // 33680 floats = 134720 B

// -------------------------------------------------------------------
// Setup: fold BN params into weights; swizzle A so one ds_load_b128
// per lane yields the WMMA B-operands of TWO adjacent j-tiles:
//   float index = kk*512 + jp*128 + L*4 + q      (q = 0..3)
//   r = q&1, j = jp*2 + (q>>1)
//   holds A[k = kk*4 + 2*(L>=16) + r][n = j*16 + L%16]
// (f32 4x16 B-operand: VGPR r, lanes 0-15 -> K=r, lanes 16-31 -> K=r+2)
// -------------------------------------------------------------------
__global__ void mlpattn_setup_kernel(const float* __restrict__ W1,
                                     const float* __restrict__ W2,
                                     const float* __restrict__ g1,
                                     const float* __restrict__ be1,
                                     const float* __restrict__ m1,
                                     const float* __restrict__ va1,
                                     const float* __restrict__ g2,
                                     const float* __restrict__ be2,
                                     const float* __restrict__ m2,
                                     const float* __restrict__ va2,
                                     float* __restrict__ ws)
{
    __shared__ float s1[384];
    __shared__ float off1[384];
    const int tid = threadIdx.x;      // 256 threads

    for (int i = tid; i < 384; i += 256) {
        float s = g1[i] * rsqrtf(va1[i] + BN_EPS);
        s1[i]   = s;
        off1[i] = be1[i] - m1[i] * s;
    }
    __syncthreads();

    for (int d = tid; d < 128; d += 256) {
        float acc = 0.f;
        for (int f = 0; f < 384; ++f) acc += off1[f] * W1[f * 128 + d];
        ws[WS_B1 + d] = acc;
        float s2 = g2[d] * rsqrtf(va2[d] + BN_EPS);
        ws[WS_V + d] = s2 * W2[d];
    }
    if (tid == 0) {
        float acc = 0.f;
        for (int d = 0; d < 128; ++d) {
            float s2 = g2[d] * rsqrtf(va2[d] + BN_EPS);
            acc += (be2[d] - m2[d] * s2) * W2[d];
        }
        ws[WS_C2] = acc;
    }

    for (int idx = tid; idx < 16384; idx += 256) {
        int q  = idx & 3;
        int L  = (idx >> 2) & 31;
        int jp = (idx >> 7) & 3;
        int kk = idx >> 9;
        int r  = q & 1;
        int j  = jp * 2 + (q >> 1);
        int k  = kk * 4 + ((L >> 4) << 1) + r;
        int n  = j * 16 + (L & 15);
        ws[WS_ASW + idx] = s1[k]       * W1[k * 128 + n]
                         + s1[128 + k] * W1[(128 + k) * 128 + n];
    }
    for (int idx = tid; idx < 16384; idx += 256) {
        int k = idx >> 7, d = idx & 127;
        ws[WS_C + idx] = s1[256 + k] * W1[(256 + k) * 128 + d]
                       - s1[128 + k] * W1[(128 + k) * 128 + d];
    }
}

#if USE_TDM
// 2D tensor-DMA load: global (tensor_d1 x tensor_d0 elems of 4B, row stride
// d0_stride) -> LDS at lds_byte_off, tile tile_d1 x tile_d0, optional LDS
// padding (pad codes per D# spec). OOB rows read as zero -> free tail fill.
__device__ __forceinline__ void tdm_load_2d_f32(unsigned lds_byte_off,
                                                const float* gaddr,
                                                unsigned tensor_d0, unsigned tensor_d1,
                                                unsigned tile_d0, unsigned tile_d1,
                                                unsigned d0_stride,
                                                unsigned pad_en,
                                                unsigned pad_interval_code,
                                                unsigned pad_amount_code)
{
    unsigned long long ga = (unsigned long long)(uintptr_t)gaddr;
    u32x4 g0;
    g0[0] = 1u;                                   // count=1, user mode
    g0[1] = lds_byte_off;
    g0[2] = (unsigned)(ga & 0xFFFFFFFFu);
    g0[3] = (unsigned)((ga >> 32) & 0x01FFFFFFu) | (2u << 30);   // type=2 (image)
    i32x8 g1;
    g1[0] = (int)((2u << 16)                      // data_size = 4B
                | (pad_en << 20)
                | (pad_interval_code << 22)
                | (pad_amount_code << 25));
    g1[1] = (int)((tensor_d0 & 0xFFFFu) << 16);   // ab_addr=0 | tensor_dim0 lo
    g1[2] = (int)(((tensor_d0 >> 16) & 0xFFFFu) | ((tensor_d1 & 0xFFFFu) << 16));
    g1[3] = (int)(((tensor_d1 >> 16) & 0xFFFFu) | ((tile_d0 & 0xFFFFu) << 16));
    g1[4] = (int)(tile_d1 & 0xFFFFu);             // tile_dim2 = 0
    g1[5] = (int)d0_stride;                       // tensor_dim0_stride lo32
    g1[6] = 0;                                    // stride0 hi | stride1 lo
    g1[7] = 0;
    i32x4 z4 = {0, 0, 0, 0};
    i32x8 z8 = {0, 0, 0, 0, 0, 0, 0, 0};
    __builtin_amdgcn_tensor_load_to_lds(g0, g1, z4, z4, z8, 0);
}
#endif

// -------------------------------------------------------------------
// Main: one block per batch element, 128 threads = 4 waves.
// -------------------------------------------------------------------
__global__ void mlpattn_main_kernel(const float* __restrict__ inputs,
                                    const float* __restrict__ query,
                                    const float* __restrict__ ws,
                                    float* __restrict__ out)
{
    extern __shared__ float sm[];
    const int tid  = threadIdx.x;     // 128
    const int lane = tid & 31;
    const int wave = tid >> 5;
    const int b    = blockIdx.x;

    sm[SM_OUT + tid] = 0.f;
    sm[SM_V + tid]   = ws[WS_V + tid];
    if (tid == 0) sm[SM_C2] = ws[WS_C2];

    const float* ibase = inputs + (size_t)b * (T_ * D_);
    float* xbuf0 = sm + SM_X + wave * (2 * TILEF);
    float* xbuf1 = xbuf0 + TILEF;
#if USE_TDM
    const unsigned ldsbase = __builtin_amdgcn_groupstaticsize();
    const unsigned xoff0 = ldsbase + (unsigned)((const char*)xbuf0 - (const char*)sm);
    const unsigned xoff1 = xoff0 + TILEF * 4u;
    // one TDM op pulls the whole 64KB swizzled-weight image into LDS
    if (wave == 0)
        tdm_load_2d_f32(ldsbase + SM_W * 4u, ws + WS_ASW,
                        16384u, 1u, 16384u, 1u, 16384u, 0u, 0u, 0u);
    // prologue: async-load this wave's first X tile (hardware pads rows to
    // XPITCH and zero-fills rows beyond T via tensor_dim1 OOB)
    if (wave < NT)
        tdm_load_2d_f32(xoff0, ibase + wave * 16 * 128,
                        128u, (unsigned)(T_ - wave * 16), 128u, 16u, 128u,
                        1u, 6u /*128 dwords*/, 3u /*4 dwords*/);
#endif

    // qc[tid] = b1'[tid] + sum_k q[b][k] * C[k][tid]   (per-batch, T-independent)
    {
        const float* q = query + (size_t)b * 128;
        const float* C = ws + WS_C;
        float acc = ws[WS_B1 + tid];
        #pragma unroll 8
        for (int k = 0; k < 128; ++k) acc += q[k] * C[k * 128 + tid];
        sm[SM_QC + tid] = acc;
    }

#if USE_TDM
    if (wave == 0) __builtin_amdgcn_s_wait_tensorcnt(1);  // weights landed
#else
    {   // weights -> LDS, manual
        const v4f* src = (const v4f*)(ws + WS_ASW);
        v4f*       dst = (v4f*)(sm + SM_W);
        #pragma unroll 4
        for (int i = tid; i < 4096; i += 128) dst[i] = src[i];
    }
#endif
    __syncthreads();

    const float c2 = sm[SM_C2];

    // tile-invariant per-lane accumulator init values (8 regs; v stays in LDS
    // to keep peak pressure under the allocator's target)
    float qcl[8];
    #pragma unroll
    for (int j = 0; j < 8; ++j)
        qcl[j] = sm[SM_QC + j * 16 + (lane & 15)];

    float colacc0 = 0.f, colacc1 = 0.f, colacc2 = 0.f, colacc3 = 0.f;
    int ib = 0;

    for (int tl = wave; tl < NT; tl += NWAVE, ib ^= 1) {
        float* xw = ib ? xbuf1 : xbuf0;
        const int tln = tl + NWAVE;

#if USE_TDM
        if (tln < NT) {
            // next tile -> other buffer; its previous readers finished long ago,
            // but fence LDS reads before the TDM overwrites the buffer.
            asm volatile("s_wait_dscnt 0x0" ::: "memory");
            tdm_load_2d_f32(ib ? xoff0 : xoff1, ibase + tln * 16 * 128,
                            128u, (unsigned)(T_ - tln * 16), 128u, 16u, 128u,
                            1u, 6u, 3u);
            __builtin_amdgcn_s_wait_tensorcnt(1);   // current tile ready
        } else {
            __builtin_amdgcn_s_wait_tensorcnt(0);
        }
#else
        const int t0 = tl * 16;
        #pragma unroll
        for (int r = 0; r < 16; ++r) {
            const int t = t0 + r;
            v4f val = {0.f, 0.f, 0.f, 0.f};
            if (t < T_) val = *(const v4f*)(ibase + t * 128 + lane * 4);
            *(v4f*)(xw + r * XPITCH + lane * 4) = val;
        }
        if (tln < NT) {
            const int tn = tln * 16;
            if ((size_t)tn * 512 + (size_t)lane * 256 < (size_t)T_ * 512)
                __builtin_prefetch((const char*)(ibase + tn * 128) + lane * 256, 0, 1);
        }
#endif

        // ---- 8 independent WMMA accumulator chains (j = output column tile)
        v8f acc[8];
        #pragma unroll
        for (int j = 0; j < 8; ++j) {
            const float qcv = qcl[j];
            v8f t = {qcv, qcv, qcv, qcv, qcv, qcv, qcv, qcv};
            acc[j] = t;
        }

        // Operand streams, both ping-pong double-buffered one kk ahead:
        //  - X A-operand: one ds_load_b64 per kk (lane<16: row=lane,K=kk*4+{0,1};
        //    lane>=16: K=kk*4+{2,3})  -- loaded per-iteration to keep live range
        //    at 2 VGPRs instead of a 64-VGPR xr[32] array (avoids spills)
        //  - weights: four ds_load_b128, each covering two j-tiles
        const float* wb = sm + SM_W + lane * 4;
        const float* xb = xw + (lane & 15) * XPITCH + ((lane >> 4) << 1);
        v4f wA[4], wB[4];
        v2f xA, xB;
        #pragma unroll
        for (int jp = 0; jp < 4; ++jp)
            wA[jp] = *(const v4f*)(wb + jp * 128);
        xA = *(const v2f*)(xb);

        #pragma unroll
        for (int kk = 0; kk < 32; ++kk) {
            const bool odd = (kk & 1) != 0;
            v4f* wc = odd ? wB : wA;
            v4f* wn = odd ? wA : wB;
            v2f  xc = odd ? xB : xA;
            if (kk < 31) {
                #pragma unroll
                for (int jp = 0; jp < 4; ++jp)
                    wn[jp] = *(const v4f*)(wb + (kk + 1) * 512 + jp * 128);
                if (odd) xA = *(const v2f*)(xb + (kk + 1) * 4);
                else     xB = *(const v2f*)(xb + (kk + 1) * 4);
            }
            SCHED_FENCE();
            #pragma unroll
            for (int jp = 0; jp < 4; ++jp) {
                v2f lo = {wc[jp][0], wc[jp][1]};
                v2f hi = {wc[jp][2], wc[jp][3]};
                acc[2 * jp] = __builtin_amdgcn_wmma_f32_16x16x4_f32(
                    false, xc, false, lo, (short)0, acc[2 * jp], false, false);
                acc[2 * jp + 1] = __builtin_amdgcn_wmma_f32_16x16x4_f32(
                    false, xc, false, hi, (short)0, acc[2 * jp + 1], false, false);
            }
            SCHED_FENCE();
        }

        // ---- relu + dot with v: lane owns column n = j*16 + lane%16
        float part[8] = {0.f, 0.f, 0.f, 0.f, 0.f, 0.f, 0.f, 0.f};
        #pragma unroll
        for (int j = 0; j < 8; ++j) {
            const float vj = sm[SM_V + j * 16 + (lane & 15)];
            #pragma unroll
            for (int r = 0; r < 8; ++r) {
                float h = fmaxf(acc[j][r], 0.f);
                part[r] += h * vj;
            }
        }
        // reduce across each 16-lane group (C/D layout: lanes 0-15 = rows 0-7, 16-31 = rows 8-15)
        #pragma unroll
        for (int r = 0; r < 8; ++r) {
            part[r] += __shfl_xor(part[r], 1, 32);
            part[r] += __shfl_xor(part[r], 2, 32);
            part[r] += __shfl_xor(part[r], 4, 32);
            part[r] += __shfl_xor(part[r], 8, 32);
        }
        float other[8];
        #pragma unroll
        for (int r = 0; r < 8; ++r) other[r] = __shfl_xor(part[r], 16, 32);

        const bool up = (lane >= 16);
        float wrow[16];
        #pragma unroll
        for (int r = 0; r < 8; ++r) {
            float llo = up ? other[r] : part[r];   // rows 0..7
            float lhi = up ? part[r] : other[r];   // rows 8..15
            wrow[r]     = 1.f / (1.f + __expf(-(llo + c2)));
            wrow[8 + r] = 1.f / (1.f + __expf(-(lhi + c2)));
        }

        // ---- out += w[row] * X[row, :]  (lane covers cols lane+{0,32,64,96};
        //      padded/OOB rows are zero so they contribute nothing)
        #pragma unroll
        for (int r = 0; r < 16; ++r) {
            const float w = wrow[r];
            const float* xrow = xw + r * XPITCH + lane;
            colacc0 += w * xrow[0];
            colacc1 += w * xrow[32];
            colacc2 += w * xrow[64];
            colacc3 += w * xrow[96];
        }
    }

    atomicAdd(&sm[SM_OUT + lane],      colacc0);
    atomicAdd(&sm[SM_OUT + lane + 32], colacc1);
    atomicAdd(&sm[SM_OUT + lane + 64], colacc2);
    atomicAdd(&sm[SM_OUT + lane + 96], colacc3);
    __syncthreads();

    out[(size_t)b * 128 + tid] = sm[SM_OUT + tid];
}

extern "C" void kernel_launch(void* const* d_in, const int* in_sizes, int n_in,
                              void* d_out, int out_size, void* d_ws, size_t ws_size,
                              hipStream_t stream) {
    (void)in_sizes; (void)n_in; (void)out_size; (void)ws_size;
    const float* inputs = (const float*)d_in[0];
    const float* query  = (const float*)d_in[1];
    const float* W1     = (const float*)d_in[2];
    const float* W2     = (const float*)d_in[3];
    const float* g1     = (const float*)d_in[4];
    const float* be1    = (const float*)d_in[5];
    const float* m1     = (const float*)d_in[6];
    const float* va1    = (const float*)d_in[7];
    const float* g2     = (const float*)d_in[8];
    const float* be2    = (const float*)d_in[9];
    const float* m2     = (const float*)d_in[10];
    const float* va2    = (const float*)d_in[11];
    float* ws  = (float*)d_ws;
    float* out = (float*)d_out;

    mlpattn_setup_kernel<<<1, 256, 0, stream>>>(W1, W2, g1, be1, m1, va1,
                                                g2, be2, m2, va2, ws);
    mlpattn_main_kernel<<<B_, 128, SM_TOTAL * sizeof(float), stream>>>(
        inputs, query, ws, out);
}